// Gate_87540023427080
// MI455X (gfx1250) — compile-verified
//
#include <hip/hip_runtime.h>
#include <math.h>

// MoE gate (DeepSeek noaux_tc style) for MI455X / gfx1250.
// Phase 1: logits = x @ W^T via V_WMMA_F32_16X16X4_F32 (fp32 tensor path),
//          fused sigmoid, scores -> d_ws.   (HBM-bound: streams 268 MB of x)
// Phase 2: per-token grouped top-k routing (tiny).

typedef __attribute__((ext_vector_type(2))) float v2f;
typedef __attribute__((ext_vector_type(8))) float v8f;

#define DIM        4096
#define NEXP       64
#define NTOK       16384
#define NGROUPS    8
#define GSIZE      8      // experts per group
#define TOPKG      4
#define TOPK       8
#define ROUTE_SCALE 2.5f

// ---------------------------------------------------------------------------
// Kernel 1: 16-token x 64-expert logits tile per block (4 waves x 16 experts),
// fp32 WMMA 16x16x4 accumulation over K=4096, fused sigmoid.
//
// Fragment mapping for V_WMMA_F32_16X16X4_F32 (ISA 7.12.2):
//   A (16x4, MxK): lanes 0-15 hold row M=lane, K={0,1} in v0,v1;
//                  lanes 16-31 hold row M=lane-16, K={2,3}.
//   B (4x16, KxN): lanes 0-15 hold col N=lane, K={0,1};
//                  lanes 16-31 hold col N=lane-16, K={2,3}.
//   Both are a contiguous float2 at [row*DIM + k + 2*(lane>>4)].
//   C (16x16):     vgpr i, lanes 0-15 -> (M=i,   N=lane),
//                            lanes 16-31 -> (M=i+8, N=lane-16).
// ---------------------------------------------------------------------------
__global__ __launch_bounds__(128) void gate_gemm_sigmoid_kernel(
    const float* __restrict__ x,       // [NTOK, DIM]
    const float* __restrict__ weight,  // [NEXP, DIM]
    float* __restrict__ scores)        // [NTOK, NEXP]
{
    const int lane  = threadIdx.x & 31;
    const int wave  = threadIdx.x >> 5;     // 0..3 -> expert tile
    const int t0    = blockIdx.x * 16;      // token tile base
    const int ebase = wave * 16;            // expert tile base
    const int r     = lane & 15;
    const int half  = lane >> 4;            // 0: K={0,1}, 1: K={2,3}

    const float* xrow = x      + (size_t)(t0 + r)    * DIM + 2 * half;
    const float* wrow = weight + (size_t)(ebase + r) * DIM + 2 * half;

    v8f c = {0.f, 0.f, 0.f, 0.f, 0.f, 0.f, 0.f, 0.f};

#pragma unroll 4
    for (int k = 0; k < DIM; k += 4) {
        v2f a = *(const v2f*)(xrow + k);
        v2f b = *(const v2f*)(wrow + k);
        // 8 args: (neg_a, A, neg_b, B, c_mod, C, reuse_a, reuse_b)
        c = __builtin_amdgcn_wmma_f32_16x16x4_f32(
                false, a, false, b, (short)0, c, false, false);
    }

    // Fused sigmoid + scatter of the 16x16 C tile.
    const int mbase = half * 8;
#pragma unroll
    for (int i = 0; i < 8; ++i) {
        float v = c[i];
        float s = 1.0f / (1.0f + __expf(-v));
        scores[(size_t)(t0 + mbase + i) * NEXP + (ebase + r)] = s;
    }
}

// ---------------------------------------------------------------------------
// Kernel 2: per-token routing. Reproduces lax.top_k tie-breaking (lowest
// index wins on equal values) via strict-greater argmax scans.
// Selection uses score+bias; returned weights use raw sigmoid scores.
// ---------------------------------------------------------------------------
__global__ __launch_bounds__(256) void gate_route_kernel(
    const float* __restrict__ scores,  // [NTOK, NEXP]
    const float* __restrict__ bias,    // [NEXP]
    float* __restrict__ w_out,         // [NTOK, TOPK]
    int*   __restrict__ idx_out)       // [NTOK, TOPK]
{
    __shared__ float sbias[NEXP];
    if (threadIdx.x < NEXP) sbias[threadIdx.x] = bias[threadIdx.x];
    __syncthreads();

    const int t = blockIdx.x * blockDim.x + threadIdx.x;
    if (t >= NTOK) return;

    float sc[NEXP];
    const float* row = scores + (size_t)t * NEXP;
#pragma unroll
    for (int e = 0; e < NEXP; ++e) sc[e] = row[e];

    // Group score = sum of top-2 (score+bias) within each group of 8.
    float gs[NGROUPS];
#pragma unroll
    for (int g = 0; g < NGROUPS; ++g) {
        float m1 = -INFINITY, m2 = -INFINITY;
#pragma unroll
        for (int j = 0; j < GSIZE; ++j) {
            float v = sc[g * GSIZE + j] + sbias[g * GSIZE + j];
            if (v > m1)      { m2 = m1; m1 = v; }
            else if (v > m2) { m2 = v; }
        }
        gs[g] = m1 + m2;
    }

    // Top-4 groups (strict > => lowest index on ties, like lax.top_k).
    unsigned gsel = 0u;
    for (int rsel = 0; rsel < TOPKG; ++rsel) {
        float best = -INFINITY; int bi = 0;
#pragma unroll
        for (int g = 0; g < NGROUPS; ++g) {
            if (!((gsel >> g) & 1u) && gs[g] > best) { best = gs[g]; bi = g; }
        }
        gsel |= 1u << bi;
    }

    // Top-8 experts among selected groups, on score+bias.
    unsigned long long taken = 0ull;
    int   idx[TOPK];
    float wts[TOPK];
    float wsum = 0.f;
    for (int rsel = 0; rsel < TOPK; ++rsel) {
        float best = -INFINITY; int bi = 0;
#pragma unroll
        for (int e = 0; e < NEXP; ++e) {
            if (((gsel >> (e >> 3)) & 1u) && !((taken >> e) & 1ull)) {
                float v = sc[e] + sbias[e];
                if (v > best) { best = v; bi = e; }
            }
        }
        taken |= 1ull << bi;
        idx[rsel] = bi;
        wts[rsel] = sc[bi];
        wsum     += sc[bi];
    }

    const float scale = ROUTE_SCALE / (wsum + 1e-6f);
#pragma unroll
    for (int rsel = 0; rsel < TOPK; ++rsel) {
        w_out[(size_t)t * TOPK + rsel]   = wts[rsel] * scale;
        idx_out[(size_t)t * TOPK + rsel] = idx[rsel];
    }
}

// ---------------------------------------------------------------------------
extern "C" void kernel_launch(void* const* d_in, const int* in_sizes, int n_in,
                              void* d_out, int out_size, void* d_ws, size_t ws_size,
                              hipStream_t stream) {
    const float* x      = (const float*)d_in[0];  // [NTOK, DIM]
    const float* weight = (const float*)d_in[1];  // [NEXP, DIM]
    const float* bias   = (const float*)d_in[2];  // [NEXP]

    float* scores = (float*)d_ws;                 // [NTOK, NEXP] = 4 MB scratch

    float* w_out  = (float*)d_out;                               // first output
    int*   i_out  = (int*)((float*)d_out + (size_t)NTOK * TOPK); // second output

    gate_gemm_sigmoid_kernel<<<NTOK / 16, 128, 0, stream>>>(x, weight, scores);
    gate_route_kernel<<<(NTOK + 255) / 256, 256, 0, stream>>>(scores, bias, w_out, i_out);
}